// Transformer_3006477107710
// MI455X (gfx1250) — compile-verified
//
#include <hip/hip_runtime.h>
#include <math.h>

typedef float v2f __attribute__((ext_vector_type(2)));
typedef float v8f __attribute__((ext_vector_type(8)));

#define CONTEXT 5
#define DIM     32
#define NTOK    13
#define HEADS   4
#define LAYERS  4

// D[16][ND] = A[16][32] @ W[32][N]  (+ bias on valid rows, optional relu)
// A, D in LDS; W in global. One wave32, EXEC must be all-ones at WMMA.
// Padded rows (m >= CONTEXT) of A are zero -> D padded rows stay zero
// (bias only added for m < CONTEXT; relu(0) == 0).
template <int N, int ND, bool RELU>
__device__ __forceinline__ void wmma_mm(const float* __restrict__ A,
                                        const float* __restrict__ W,
                                        float* __restrict__ D,
                                        const float* __restrict__ bias,
                                        int lane)
{
    const int m_a   = lane & 15;            // A row for this lane
    const int khalf = (lane >> 4) << 1;     // 0 (lanes 0-15) or 2 (lanes 16-31)
    const int ncol  = lane & 15;            // B/D column for this lane
    const int mhi   = (lane >> 4) << 3;     // D row offset: 0 or 8
    constexpr int NT = (N + 15) >> 4;

#pragma unroll
    for (int nt = 0; nt < NT; ++nt) {
        v8f acc = {};
        const int  n      = nt * 16 + ncol;
        const bool nvalid = (n < N);
#pragma unroll
        for (int kk = 0; kk < 8; ++kk) {     // K = 32 in steps of 4
            const int kb = kk * 4 + khalf;
            v2f a, b;
            a.x = A[m_a * 32 + kb];
            a.y = A[m_a * 32 + kb + 1];
            b.x = nvalid ? W[kb * N + n]       : 0.0f;
            b.y = nvalid ? W[(kb + 1) * N + n] : 0.0f;
            // (neg_a, A, neg_b, B, c_mod, C, reuse_a, reuse_b)
            acc = __builtin_amdgcn_wmma_f32_16x16x4_f32(
                false, a, false, b, (short)0, acc, false, false);
        }
        const float bv = (bias != nullptr && nvalid) ? bias[n] : 0.0f;
#pragma unroll
        for (int r = 0; r < 8; ++r) {
            const int m = r + mhi;
            float v = acc[r];
            if (m < CONTEXT) v += bv;        // keep padded rows exactly zero
            if (RELU) v = fmaxf(v, 0.0f);
            if (nvalid) D[m * ND + n] = v;
        }
    }
}

__global__ __launch_bounds__(32, 1) void tf_fwd_kernel(
    const int*   __restrict__ toks,
    const float* __restrict__ pos_embed,   // [5][32]
    const float* __restrict__ tok_embed,   // [13][32]
    const float* __restrict__ Wq,          // [4][32][32]
    const float* __restrict__ Wk,
    const float* __restrict__ Wv,
    const float* __restrict__ Wp,
    const float* __restrict__ bp,          // [4][32]
    const float* __restrict__ W1,
    const float* __restrict__ b1,
    const float* __restrict__ W2,
    const float* __restrict__ b2,
    const float* __restrict__ Wu,          // [32][13]
    const float* __restrict__ bu,          // [13]
    float*       __restrict__ out)         // [5][13]
{
    __shared__ float Xs[16 * 32];
    __shared__ float Qs[16 * 32];
    __shared__ float Ks[16 * 32];
    __shared__ float Vs[16 * 32];
    __shared__ float Yc[16 * 32];
    __shared__ float Yp[16 * 32];
    __shared__ float Hs[16 * 32];
    __shared__ float Ms[16 * 32];
    __shared__ float Ss[HEADS][CONTEXT][CONTEXT];
    __shared__ float Ob[16 * 16];

    const int lane = threadIdx.x;

    // Zero the buffers whose padded rows must stay zero across the network.
    for (int i = lane; i < 16 * 32; i += 32) { Xs[i] = 0.0f; Yc[i] = 0.0f; }
    __syncthreads();

    // Embedding: X[c] = tok_embed[toks[c]] + pos_embed[c]   (rows 5..15 stay 0)
    for (int c = 0; c < CONTEXT; ++c) {
        const int t = toks[c];
        Xs[c * 32 + lane] = tok_embed[t * DIM + lane] + pos_embed[c * DIM + lane];
    }
    __syncthreads();

    const float inv_sqrt_dim = 0.17677669529663687f;   // 1/sqrt(32)

    for (int l = 0; l < LAYERS; ++l) {
        const float* wq  = Wq + l * DIM * DIM;
        const float* wk  = Wk + l * DIM * DIM;
        const float* wv  = Wv + l * DIM * DIM;
        const float* wp  = Wp + l * DIM * DIM;
        const float* w1  = W1 + l * DIM * DIM;
        const float* w2  = W2 + l * DIM * DIM;
        const float* bpl = bp + l * DIM;
        const float* b1l = b1 + l * DIM;
        const float* b2l = b2 + l * DIM;

        // Prefetch next layer's weights into cache while we compute this one.
        if (l + 1 < LAYERS) {
            const long o = (long)(l + 1) * DIM * DIM + (long)lane * 32; // 128B/lane
            __builtin_prefetch(Wq + o, 0, 0);
            __builtin_prefetch(Wk + o, 0, 0);
            __builtin_prefetch(Wv + o, 0, 0);
            __builtin_prefetch(Wp + o, 0, 0);
            __builtin_prefetch(W1 + o, 0, 0);
            __builtin_prefetch(W2 + o, 0, 0);
        }

        // Q, K, V projections (WMMA)
        wmma_mm<32, 32, false>(Xs, wq, Qs, nullptr, lane);
        wmma_mm<32, 32, false>(Xs, wk, Ks, nullptr, lane);
        wmma_mm<32, 32, false>(Xs, wv, Vs, nullptr, lane);
        __syncthreads();

        // Causal attention scores + softmax: lane = h*5 + q  (20 active lanes)
        if (lane < HEADS * CONTEXT) {
            const int h = lane / CONTEXT;
            const int q = lane % CONTEXT;
            float s[CONTEXT];
#pragma unroll
            for (int k = 0; k < CONTEXT; ++k) {
                float d = 0.0f;
#pragma unroll
                for (int e = 0; e < 8; ++e)
                    d += Qs[q * 32 + h * 8 + e] * Ks[k * 32 + h * 8 + e];
                s[k] = (k <= q) ? d * inv_sqrt_dim : -__builtin_huge_valf();
            }
            float mx = s[0];
#pragma unroll
            for (int k = 1; k < CONTEXT; ++k) mx = fmaxf(mx, s[k]);
            float ex[CONTEXT], sum = 0.0f;
#pragma unroll
            for (int k = 0; k < CONTEXT; ++k) { ex[k] = expf(s[k] - mx); sum += ex[k]; }
            const float inv = 1.0f / sum;
#pragma unroll
            for (int k = 0; k < CONTEXT; ++k) Ss[h][q][k] = ex[k] * inv;
        }
        __syncthreads();

        // Ycat[q][c] = sum_k S[h(c)][q][k] * V[k][c]   (rows 5..15 stay 0)
        for (int q = 0; q < CONTEXT; ++q) {
            const int h = lane >> 3;
            float y = 0.0f;
            for (int k = 0; k <= q; ++k)
                y += Ss[h][q][k] * Vs[k * 32 + lane];
            Yc[q * 32 + lane] = y;
        }
        __syncthreads();

        // Output projection + MLP (WMMA)
        wmma_mm<32, 32, false>(Yc, wp, Yp, bpl, lane);     // Yproj
        __syncthreads();
        wmma_mm<32, 32, true >(Yp, w1, Hs, b1l, lane);     // relu(Yproj@W1+b1)
        __syncthreads();
        wmma_mm<32, 32, false>(Hs, w2, Ms, b2l, lane);     // H@W2+b2
        __syncthreads();

        // Residuals: X = (Y + Yproj) + X
        for (int c = 0; c < CONTEXT; ++c)
            Xs[c * 32 + lane] += Ms[c * 32 + lane] + Yp[c * 32 + lane];
        __syncthreads();
    }

    // Unembed: logits = X @ Wu + bu   (N = 13, one 16-wide tile)
    wmma_mm<NTOK, 16, false>(Xs, Wu, Ob, nullptr, lane);
    __syncthreads();
    for (int i = lane; i < CONTEXT * NTOK; i += 32) {
        const int q = i / NTOK;
        const int t = i % NTOK;
        out[i] = Ob[q * 16 + t] + bu[t];
    }
}

extern "C" void kernel_launch(void* const* d_in, const int* in_sizes, int n_in,
                              void* d_out, int out_size, void* d_ws, size_t ws_size,
                              hipStream_t stream) {
    (void)in_sizes; (void)n_in; (void)out_size; (void)d_ws; (void)ws_size;
    const int*   toks      = (const int*)  d_in[0];
    const float* pos_embed = (const float*)d_in[1];
    const float* tok_embed = (const float*)d_in[2];
    const float* Wq        = (const float*)d_in[3];
    const float* Wk        = (const float*)d_in[4];
    const float* Wv        = (const float*)d_in[5];
    const float* Wp        = (const float*)d_in[6];
    const float* bp        = (const float*)d_in[7];
    const float* W1        = (const float*)d_in[8];
    const float* b1        = (const float*)d_in[9];
    const float* W2        = (const float*)d_in[10];
    const float* b2        = (const float*)d_in[11];
    const float* Wu        = (const float*)d_in[12];
    const float* bu        = (const float*)d_in[13];
    float*       out       = (float*)d_out;

    tf_fwd_kernel<<<1, 32, 0, stream>>>(toks, pos_embed, tok_embed,
                                        Wq, Wk, Wv, Wp, bp,
                                        W1, b1, W2, b2, Wu, bu, out);
}